// HGNNScheduler_33921651704176
// MI455X (gfx1250) — compile-verified
//
#include <hip/hip_runtime.h>
#include <hip/hip_bf16.h>
#include <math.h>

#define Bc 32
#define Oc 1000
#define Mc 20
#define Jc 100
#define OUTc 8
#define Hc 2
#define Dc 4

typedef float v2f __attribute__((ext_vector_type(2)));
typedef float v8f __attribute__((ext_vector_type(8)));

// ---------------------------------------------------------------------------
// Normalization kernels
// ---------------------------------------------------------------------------
__global__ __launch_bounds__(256) void norm_opes_kernel(const float* __restrict__ raw,
                                                        float* __restrict__ out) {
  int b = blockIdx.x, tid = threadIdx.x;
  float s[4] = {0.f, 0.f, 0.f, 0.f}, q[4] = {0.f, 0.f, 0.f, 0.f};
  for (int o = tid; o < Oc; o += 256) {
    const float* r = &raw[(b * Oc + o) * 4];
    #pragma unroll
    for (int f = 0; f < 4; ++f) { float v = r[f]; s[f] += v; q[f] += v * v; }
  }
  __shared__ float rs[4][256], rq[4][256];
  #pragma unroll
  for (int f = 0; f < 4; ++f) { rs[f][tid] = s[f]; rq[f][tid] = q[f]; }
  __syncthreads();
  for (int st = 128; st > 0; st >>= 1) {
    if (tid < st) {
      #pragma unroll
      for (int f = 0; f < 4; ++f) { rs[f][tid] += rs[f][tid + st]; rq[f][tid] += rq[f][tid + st]; }
    }
    __syncthreads();
  }
  __shared__ float meanv[4], stdv[4];
  if (tid < 4) {
    float mu = rs[tid][0] / (float)Oc;
    float var = (rq[tid][0] - rs[tid][0] * rs[tid][0] / (float)Oc) / (float)(Oc - 1);
    meanv[tid] = mu;
    stdv[tid] = sqrtf(fmaxf(var, 0.f)) + 1e-5f;
  }
  __syncthreads();
  for (int o = tid; o < Oc; o += 256) {
    const float* r = &raw[(b * Oc + o) * 4];
    float* w = &out[(b * Oc + o) * 4];
    #pragma unroll
    for (int f = 0; f < 4; ++f) w[f] = (r[f] - meanv[f]) / stdv[f];
  }
}

__global__ void norm_mas_kernel(const float* __restrict__ raw, float* __restrict__ out) {
  int b = blockIdx.x * blockDim.x + threadIdx.x;
  if (b >= Bc) return;
  for (int f = 0; f < 2; ++f) {
    float s = 0.f, q = 0.f;
    for (int m = 0; m < Mc; ++m) { float v = raw[(b * Mc + m) * 2 + f]; s += v; q += v * v; }
    float mu = s / (float)Mc;
    float var = (q - s * s / (float)Mc) / (float)(Mc - 1);
    float sd = sqrtf(fmaxf(var, 0.f)) + 1e-5f;
    for (int m = 0; m < Mc; ++m)
      out[(b * Mc + m) * 2 + f] = (raw[(b * Mc + m) * 2 + f] - mu) / sd;
  }
}

__global__ __launch_bounds__(256) void proc_partial_kernel(const float* __restrict__ x,
                                                           float* __restrict__ part) {
  const int N = Bc * Oc * Mc;
  int tid = threadIdx.x;
  float s = 0.f, q = 0.f;
  for (int i = blockIdx.x * 256 + tid; i < N; i += gridDim.x * 256) {
    float v = x[i]; s += v; q += v * v;
  }
  __shared__ float rs[256], rq[256];
  rs[tid] = s; rq[tid] = q;
  __syncthreads();
  for (int st = 128; st > 0; st >>= 1) {
    if (tid < st) { rs[tid] += rs[tid + st]; rq[tid] += rq[tid + st]; }
    __syncthreads();
  }
  if (tid == 0) { part[blockIdx.x] = rs[0]; part[256 + blockIdx.x] = rq[0]; }
}

__global__ void proc_final_kernel(const float* __restrict__ part, float* __restrict__ st) {
  if (threadIdx.x == 0 && blockIdx.x == 0) {
    float s = 0.f, q = 0.f;
    for (int i = 0; i < 256; ++i) { s += part[i]; q += part[256 + i]; }  // fixed order: deterministic
    const float N = (float)(Bc * Oc * Mc);
    float mu = s / N;
    float var = (q - s * s / N) / (N - 1.f);
    st[0] = s; st[1] = q; st[2] = mu; st[3] = sqrtf(fmaxf(var, 0.f)) + 1e-5f;
  }
}

// ---------------------------------------------------------------------------
// GAT layer
// ---------------------------------------------------------------------------
__global__ void gat_proj_opes_kernel(const float* __restrict__ x, int din,
                                     const float* __restrict__ Wo, const float* __restrict__ al,
                                     float* __restrict__ ho, float* __restrict__ el) {
  int i = blockIdx.x * blockDim.x + threadIdx.x;
  if (i >= Bc * Oc) return;
  float xi[8];
  for (int k = 0; k < din; ++k) xi[k] = x[i * din + k];
  float e = 0.f;
  #pragma unroll
  for (int n = 0; n < 8; ++n) {
    float s = 0.f;
    for (int k = 0; k < din; ++k) s += xi[k] * Wo[k * 8 + n];
    ho[i * 8 + n] = s;
    e += s * al[n];
  }
  el[i] = e;
}

__global__ void gat_proj_mas_kernel(const float* __restrict__ x, int din,
                                    const float* __restrict__ Wm, const float* __restrict__ ar,
                                    float* __restrict__ hm, float* __restrict__ er) {
  int i = blockIdx.x * blockDim.x + threadIdx.x;
  if (i >= Bc * Mc) return;
  float xi[8];
  for (int k = 0; k < din; ++k) xi[k] = x[i * din + k];
  float e = 0.f;
  #pragma unroll
  for (int n = 0; n < 8; ++n) {
    float s = 0.f;
    for (int k = 0; k < din; ++k) s += xi[k] * Wm[k * 8 + n];
    hm[i * 8 + n] = s;
    e += s * ar[n];
  }
  er[i] = e;
}

__global__ __launch_bounds__(256) void gat_attn_kernel(
    const float* __restrict__ ho, const float* __restrict__ el,
    const float* __restrict__ hm, const float* __restrict__ er,
    const float* __restrict__ proc, const float* __restrict__ pstats,
    const int* __restrict__ adj,
    const float* __restrict__ we, const float* __restrict__ ae,
    float* __restrict__ hmas_out) {
  int bm = blockIdx.x;
  int b = bm / Mc, m = bm % Mc;
  int tid = threadIdx.x;
  float s = 0.f;
  #pragma unroll
  for (int i = 0; i < 8; ++i) s += we[i] * ae[i];
  float mean = pstats[2], stdp = pstats[3];
  float erm = er[b * Mc + m];
  float eself = 2.f * erm; eself = eself > 0.f ? eself : 0.2f * eself;
  float mx = eself;
  for (int o = tid; o < Oc; o += 256) {
    float pn = (proc[(b * Oc + o) * Mc + m] - mean) / stdp;
    float e = el[b * Oc + o] + s * pn + erm;
    e = e > 0.f ? e : 0.2f * e;
    if (adj[(b * Oc + o) * Mc + m] != 1) e = -1e9f;
    mx = fmaxf(mx, e);
  }
  __shared__ float red[256];
  red[tid] = mx;
  __syncthreads();
  for (int st = 128; st > 0; st >>= 1) {
    if (tid < st) red[tid] = fmaxf(red[tid], red[tid + st]);
    __syncthreads();
  }
  mx = red[0];
  __syncthreads();
  float wsum = 0.f, acc[8] = {0.f, 0.f, 0.f, 0.f, 0.f, 0.f, 0.f, 0.f};
  for (int o = tid; o < Oc; o += 256) {
    float pn = (proc[(b * Oc + o) * Mc + m] - mean) / stdp;
    float e = el[b * Oc + o] + s * pn + erm;
    e = e > 0.f ? e : 0.2f * e;
    if (adj[(b * Oc + o) * Mc + m] != 1) e = -1e9f;
    float wgt = expf(e - mx);
    wsum += wgt;
    const float* h = &ho[(b * Oc + o) * 8];
    #pragma unroll
    for (int d = 0; d < 8; ++d) acc[d] += wgt * h[d];
  }
  __shared__ float r9[9][256];
  r9[0][tid] = wsum;
  #pragma unroll
  for (int d = 0; d < 8; ++d) r9[d + 1][tid] = acc[d];
  __syncthreads();
  for (int st = 128; st > 0; st >>= 1) {
    if (tid < st) {
      #pragma unroll
      for (int q2 = 0; q2 < 9; ++q2) r9[q2][tid] += r9[q2][tid + st];
    }
    __syncthreads();
  }
  if (tid < 8) {
    float wself = expf(eself - mx);
    float tot = r9[0][0] + wself;
    float o8 = (r9[tid + 1][0] + wself * hm[(b * Mc + m) * 8 + tid]) / tot;
    hmas_out[(b * Mc + m) * 8 + tid] = 1.f / (1.f + expf(-o8));
  }
}

// ---------------------------------------------------------------------------
// HGT layer
// ---------------------------------------------------------------------------
__global__ void hgt_proj_opes_kernel(const float* __restrict__ x, int din,
                                     const float* __restrict__ Wq, const float* __restrict__ Wko,
                                     const float* __restrict__ Wvo, const float* __restrict__ Wsk,
                                     float* __restrict__ q, float* __restrict__ ko,
                                     float* __restrict__ vo, float* __restrict__ sk) {
  int i = blockIdx.x * blockDim.x + threadIdx.x;
  if (i >= Bc * Oc) return;
  float xi[8];
  for (int k = 0; k < din; ++k) xi[k] = x[i * din + k];
  #pragma unroll
  for (int n = 0; n < 8; ++n) {
    float sq = 0.f, s1 = 0.f, s2 = 0.f, s3 = 0.f;
    for (int k = 0; k < din; ++k) {
      float v = xi[k];
      sq += v * Wq[k * 8 + n]; s1 += v * Wko[k * 8 + n];
      s2 += v * Wvo[k * 8 + n]; s3 += v * Wsk[k * 8 + n];
    }
    q[i * 8 + n] = sq; ko[i * 8 + n] = s1; vo[i * 8 + n] = s2; sk[i * 8 + n] = s3;
  }
}

__global__ void hgt_proj_mas_kernel(const float* __restrict__ x,
                                    const float* __restrict__ Wkm, const float* __restrict__ Wvm,
                                    float* __restrict__ km, float* __restrict__ vm) {
  int i = blockIdx.x * blockDim.x + threadIdx.x;
  if (i >= Bc * Mc) return;
  float xi[8];
  #pragma unroll
  for (int k = 0; k < 8; ++k) xi[k] = x[i * 8 + k];
  #pragma unroll
  for (int n = 0; n < 8; ++n) {
    float s1 = 0.f, s2 = 0.f;
    #pragma unroll
    for (int k = 0; k < 8; ++k) { s1 += xi[k] * Wkm[k * 8 + n]; s2 += xi[k] * Wvm[k * 8 + n]; }
    km[i * 8 + n] = s1; vm[i * 8 + n] = s2;
  }
}

// one wave32 per (b, o, head): fused masked scores + online softmax + message
__global__ __launch_bounds__(256) void hgt_attn_kernel(
    const float* __restrict__ q, const float* __restrict__ km, const float* __restrict__ vm,
    const float* __restrict__ ko, const float* __restrict__ vo,
    const int* __restrict__ adj_om, const int* __restrict__ adj_oo,
    float* __restrict__ msg) {
  int gw = (int)((blockIdx.x * blockDim.x + threadIdx.x) >> 5);
  int lane = threadIdx.x & 31;
  if (gw >= Bc * Oc * Hc) return;
  int b = gw / (Oc * Hc);
  int rem = gw % (Oc * Hc);
  int o = rem / Hc, h = rem % Hc;
  const float* qp = &q[(b * Oc + o) * 8 + h * 4];
  float q0 = qp[0], q1 = qp[1], q2 = qp[2], q3 = qp[3];
  const int* am = &adj_om[(b * Oc + o) * Mc];
  const int* ao = &adj_oo[(b * Oc + o) * Oc];
  float mx = -INFINITY, sm = 0.f, a0 = 0.f, a1 = 0.f, a2 = 0.f, a3 = 0.f;
  for (int k = lane; k < Mc + Oc; k += 32) {
    const float* kv; const float* vv; int mk;
    if (k < Mc) { mk = am[k]; kv = &km[(b * Mc + k) * 8 + h * 4]; vv = &vm[(b * Mc + k) * 8 + h * 4]; }
    else { int p = k - Mc; mk = ao[p]; kv = &ko[(b * Oc + p) * 8 + h * 4]; vv = &vo[(b * Oc + p) * 8 + h * 4]; }
    float sc2 = (mk == 1) ? 0.5f * (q0 * kv[0] + q1 * kv[1] + q2 * kv[2] + q3 * kv[3]) : -1e9f;
    float nm = fmaxf(mx, sc2);
    float f = expf(mx - nm);     // mx = -inf on first iter -> 0
    float w = expf(sc2 - nm);
    sm = sm * f + w;
    a0 = a0 * f + w * vv[0];
    a1 = a1 * f + w * vv[1];
    a2 = a2 * f + w * vv[2];
    a3 = a3 * f + w * vv[3];
    mx = nm;
  }
  for (int off = 16; off > 0; off >>= 1) {
    float mo = __shfl_xor(mx, off, 32);
    float so = __shfl_xor(sm, off, 32);
    float b0 = __shfl_xor(a0, off, 32);
    float b1v = __shfl_xor(a1, off, 32);
    float b2v = __shfl_xor(a2, off, 32);
    float b3v = __shfl_xor(a3, off, 32);
    float nm = fmaxf(mx, mo);
    float f1 = expf(mx - nm), f2 = expf(mo - nm);
    sm = sm * f1 + so * f2;
    a0 = a0 * f1 + b0 * f2; a1 = a1 * f1 + b1v * f2;
    a2 = a2 * f1 + b2v * f2; a3 = a3 * f1 + b3v * f2;
    mx = nm;
  }
  if (lane == 0) {
    float inv = 1.f / sm;
    float* op = &msg[(b * Oc + o) * 8 + h * 4];
    op[0] = a0 * inv; op[1] = a1 * inv; op[2] = a2 * inv; op[3] = a3 * inv;
  }
}

__global__ void hgt_out_kernel(const float* __restrict__ msg, const float* __restrict__ sk,
                               const float* __restrict__ Wout, float* __restrict__ out) {
  int i = blockIdx.x * blockDim.x + threadIdx.x;
  if (i >= Bc * Oc) return;
  float mi[8];
  #pragma unroll
  for (int k = 0; k < 8; ++k) mi[k] = msg[i * 8 + k];
  #pragma unroll
  for (int n = 0; n < 8; ++n) {
    float s = sk[i * 8 + n];
    #pragma unroll
    for (int k = 0; k < 8; ++k) s += mi[k] * Wout[k * 8 + n];
    out[i * 8 + n] = s > 0.f ? s : expm1f(s);   // elu
  }
}

// ---------------------------------------------------------------------------
// Pools + head MLPs
// ---------------------------------------------------------------------------
__global__ __launch_bounds__(256) void pool_opes_kernel(const float* __restrict__ hopes,
                                                        float* __restrict__ op0) {
  int b = blockIdx.x, tid = threadIdx.x;
  int f = tid & 7, g = tid >> 3;
  float acc = 0.f;
  for (int o = g; o < Oc; o += 32) acc += hopes[(b * Oc + o) * 8 + f];
  __shared__ float red[256];
  red[tid] = acc;
  __syncthreads();
  for (int st = 16; st > 0; st >>= 1) {
    if (g < st) red[tid] += red[tid + st * 8];
    __syncthreads();
  }
  if (tid < 8) op0[b * 8 + tid] = red[tid] * (1.f / (float)Oc);
}

__global__ void pool_mas_kernel(const float* __restrict__ hmas, float* __restrict__ ma0) {
  int i = blockIdx.x * blockDim.x + threadIdx.x;
  if (i >= Bc * 8) return;
  int b = i >> 3, f = i & 7;
  float s = 0.f;
  for (int m = 0; m < Mc; ++m) s += hmas[(b * Mc + m) * 8 + f];
  ma0[i] = s * (1.f / (float)Mc);
}

__global__ __launch_bounds__(64) void head_kernel(
    const float* __restrict__ scale_in, const float* __restrict__ op0, const float* __restrict__ ma0,
    const float* __restrict__ c1w1, const float* __restrict__ c1w2,
    const float* __restrict__ pw1, const float* __restrict__ pw2,
    float* __restrict__ op_pool, float* __restrict__ ma_pool) {
  int b = blockIdx.x, t = threadIdx.x;
  __shared__ float h1[64];
  __shared__ float se[8];
  float a = 0.f;
  #pragma unroll
  for (int k = 0; k < 4; ++k) a += scale_in[b * 4 + k] * c1w1[k * 64 + t];
  h1[t] = fmaxf(a, 0.f);
  __syncthreads();
  if (t < 8) {
    float s2 = 0.f;
    for (int k = 0; k < 64; ++k) s2 += h1[k] * c1w2[k * 8 + t];
    se[t] = s2;
  }
  __syncthreads();
  float a2 = 0.f;
  #pragma unroll
  for (int k = 0; k < 8; ++k) a2 += (op0[b * 8 + k] + se[k]) * pw1[k * 64 + t];
  h1[t] = fmaxf(a2, 0.f);
  __syncthreads();
  if (t < 8) {
    float s2 = 0.f;
    for (int k = 0; k < 64; ++k) s2 += h1[k] * pw2[k * 8 + t];
    op_pool[b * 8 + t] = op0[b * 8 + t] + s2;
  }
  __syncthreads();
  float a3 = 0.f;
  #pragma unroll
  for (int k = 0; k < 8; ++k) a3 += (ma0[b * 8 + k] + se[k]) * pw1[k * 64 + t];
  h1[t] = fmaxf(a3, 0.f);
  __syncthreads();
  if (t < 8) {
    float s2 = 0.f;
    for (int k = 0; k < 64; ++k) s2 += h1[k] * pw2[k * 8 + t];
    ma_pool[b * 8 + t] = ma0[b * 8 + t] + s2;
  }
}

// ---------------------------------------------------------------------------
// Action MLP: X[64000,32] @W1 -> tanh -> @W2 -> tanh -> .w3  using f32 WMMA
// Block = 128 threads (4 waves), 16 rows per block; each wave owns a 16x16 tile.
// V_WMMA_F32_16X16X4_F32 chained over K in steps of 4 (full f32 precision).
// ---------------------------------------------------------------------------
__global__ __launch_bounds__(128) void action_mlp_kernel(
    const float* __restrict__ hopes, const float* __restrict__ hmas,
    const float* __restrict__ op_pool, const float* __restrict__ ma_pool,
    const int* __restrict__ jobs,
    const float* __restrict__ W1, const float* __restrict__ b1,
    const float* __restrict__ W2, const float* __restrict__ b2,
    const float* __restrict__ w3, const float* __restrict__ b3,
    float* __restrict__ scores) {
  __shared__ float X0[16][33];
  __shared__ float X1[16][65];
  __shared__ float X2[16][65];
  __shared__ float red[16][8];
  int tile = blockIdx.x;
  int tid = threadIdx.x;
  // assemble 16 rows of h_actions = [h_jobs | h_mas | op_pool | ma_pool]
  for (int idx = tid; idx < 16 * 32; idx += 128) {
    int r = idx >> 5, c = idx & 31;
    int g = tile * 16 + r;
    int b = g / (Mc * Jc);
    int rem = g % (Mc * Jc);
    int m = rem / Jc, j = rem % Jc;
    float v;
    if (c < 8)       { int o = jobs[b * Jc + j]; v = hopes[(b * Oc + o) * 8 + c]; }
    else if (c < 16) { v = hmas[(b * Mc + m) * 8 + (c - 8)]; }
    else if (c < 24) { v = op_pool[b * 8 + (c - 16)]; }
    else             { v = ma_pool[b * 8 + (c - 24)]; }
    X0[r][c] = v;
  }
  __syncthreads();
  int wave = tid >> 5, lane = tid & 31;
  int row = lane & 15;            // A: M = lane%16
  int kk = (lane >> 4) << 1;      // A/B: K = vgpr + 2*(lane>=16)
  int ncol = (wave << 4) + row;   // B/D: N = lane%16 (+wave tile)
  int rbase = (lane >> 4) * 8;    // D: M = vgpr + 8*(lane>=16)

  // Layer 1: K = 32 -> 8 wmma steps
  v8f acc = {0.f, 0.f, 0.f, 0.f, 0.f, 0.f, 0.f, 0.f};
  #pragma unroll
  for (int s = 0; s < 8; ++s) {
    int k0 = s * 4;
    v2f av, bv;
    av.x = X0[row][k0 + kk];
    av.y = X0[row][k0 + kk + 1];
    bv.x = W1[(k0 + kk) * 64 + ncol];
    bv.y = W1[(k0 + kk + 1) * 64 + ncol];
    acc = __builtin_amdgcn_wmma_f32_16x16x4_f32(false, av, false, bv, (short)0, acc, false, false);
  }
  float bias1 = b1[ncol];
  #pragma unroll
  for (int i = 0; i < 8; ++i) X1[rbase + i][ncol] = tanhf(acc[i] + bias1);
  __syncthreads();

  // Layer 2: K = 64 -> 16 wmma steps
  v8f acc2 = {0.f, 0.f, 0.f, 0.f, 0.f, 0.f, 0.f, 0.f};
  #pragma unroll
  for (int s = 0; s < 16; ++s) {
    int k0 = s * 4;
    v2f av, bv;
    av.x = X1[row][k0 + kk];
    av.y = X1[row][k0 + kk + 1];
    bv.x = W2[(k0 + kk) * 64 + ncol];
    bv.y = W2[(k0 + kk + 1) * 64 + ncol];
    acc2 = __builtin_amdgcn_wmma_f32_16x16x4_f32(false, av, false, bv, (short)0, acc2, false, false);
  }
  float bias2 = b2[ncol];
  #pragma unroll
  for (int i = 0; i < 8; ++i) X2[rbase + i][ncol] = tanhf(acc2[i] + bias2);
  __syncthreads();

  // Layer 3: dot with w3[64]
  int r3 = tid >> 3, p3 = tid & 7;
  float s3 = 0.f;
  #pragma unroll
  for (int n = p3 * 8; n < p3 * 8 + 8; ++n) s3 += X2[r3][n] * w3[n];
  red[r3][p3] = s3;
  __syncthreads();
  if (tid < 16) {
    float s = b3[0];
    #pragma unroll
    for (int p = 0; p < 8; ++p) s += red[tid][p];
    scores[tile * 16 + tid] = s;
  }
}

// ---------------------------------------------------------------------------
// Masked softmax over M*J = 2000 per batch
// ---------------------------------------------------------------------------
__global__ __launch_bounds__(256) void softmax_kernel(const float* __restrict__ scores,
                                                      const unsigned char* __restrict__ elig,
                                                      float* __restrict__ out) {
  __shared__ float sc[Mc * Jc];
  __shared__ float red[256];
  int b = blockIdx.x, tid = threadIdx.x;
  for (int i = tid; i < Mc * Jc; i += 256) {
    int m = i / Jc, j = i % Jc;
    float v = scores[b * Mc * Jc + i];
    if (elig[(b * Jc + j) * Mc + m] == 0) v = -1e9f;
    sc[i] = v;
  }
  __syncthreads();
  float mx = -INFINITY;
  for (int i = tid; i < Mc * Jc; i += 256) mx = fmaxf(mx, sc[i]);
  red[tid] = mx;
  __syncthreads();
  for (int st = 128; st > 0; st >>= 1) {
    if (tid < st) red[tid] = fmaxf(red[tid], red[tid + st]);
    __syncthreads();
  }
  mx = red[0];
  __syncthreads();
  float sm = 0.f;
  for (int i = tid; i < Mc * Jc; i += 256) sm += expf(sc[i] - mx);
  red[tid] = sm;
  __syncthreads();
  for (int st = 128; st > 0; st >>= 1) {
    if (tid < st) red[tid] += red[tid + st];
    __syncthreads();
  }
  float inv = 1.f / red[0];
  for (int i = tid; i < Mc * Jc; i += 256) out[b * Mc * Jc + i] = expf(sc[i] - mx) * inv;
}

// ---------------------------------------------------------------------------
// Host launch
// ---------------------------------------------------------------------------
extern "C" void kernel_launch(void* const* d_in, const int* in_sizes, int n_in,
                              void* d_out, int out_size, void* d_ws, size_t ws_size,
                              hipStream_t stream) {
  (void)in_sizes; (void)n_in; (void)out_size; (void)ws_size;
  const float* raw_opes = (const float*)d_in[0];
  const float* raw_mas  = (const float*)d_in[1];
  const float* proc     = (const float*)d_in[2];
  const float* scale_in = (const float*)d_in[3];
  const int* adj_om     = (const int*)d_in[4];
  const int* adj_oo     = (const int*)d_in[5];
  const int* jobs       = (const int*)d_in[6];
  const unsigned char* elig = (const unsigned char*)d_in[7];  // jax bool = 1 byte
  const float* P[36];
  for (int i = 0; i < 36; ++i) P[i] = (const float*)d_in[8 + i];
  const float* gat_Wo[2] = {P[0], P[6]};
  const float* gat_Wm[2] = {P[1], P[7]};
  const float* gat_we[2] = {P[2], P[8]};
  const float* gat_al[2] = {P[3], P[9]};
  const float* gat_ar[2] = {P[4], P[10]};
  const float* gat_ae[2] = {P[5], P[11]};
  const float* hgt_Wq[2]   = {P[12], P[19]};
  const float* hgt_Wkm[2]  = {P[13], P[20]};
  const float* hgt_Wvm[2]  = {P[14], P[21]};
  const float* hgt_Wko[2]  = {P[15], P[22]};
  const float* hgt_Wvo[2]  = {P[16], P[23]};
  const float* hgt_Wout[2] = {P[17], P[24]};
  const float* hgt_Wskip[2] = {P[18], P[25]};
  const float* c1w1 = P[26]; const float* c1w2 = P[27];
  const float* pw1 = P[28];  const float* pw2 = P[29];
  const float* aw1 = P[30];  const float* ab1 = P[31];
  const float* aw2 = P[32];  const float* ab2 = P[33];
  const float* aw3 = P[34];  const float* ab3 = P[35];

  float* w = (float*)d_ws;
  size_t off = 0;
  auto alloc = [&](size_t n) { float* p = w + off; off += n; return p; };
  float* opes_n  = alloc((size_t)Bc * Oc * 4);
  float* mas_n   = alloc((size_t)Bc * Mc * 2);
  float* pstats  = alloc(4);
  float* partial = alloc(512);
  float* hopesA  = alloc((size_t)Bc * Oc * 8);
  float* hopesB  = alloc((size_t)Bc * Oc * 8);
  float* hmasA   = alloc((size_t)Bc * Mc * 8);
  float* hmasB   = alloc((size_t)Bc * Mc * 8);
  float* t_ho    = alloc((size_t)Bc * Oc * 8);
  float* t_el    = alloc((size_t)Bc * Oc);
  float* t_hm    = alloc((size_t)Bc * Mc * 8);
  float* t_er    = alloc((size_t)Bc * Mc);
  float* t_q     = alloc((size_t)Bc * Oc * 8);
  float* t_ko    = alloc((size_t)Bc * Oc * 8);
  float* t_vo    = alloc((size_t)Bc * Oc * 8);
  float* t_skip  = alloc((size_t)Bc * Oc * 8);
  float* t_km    = alloc((size_t)Bc * Mc * 8);
  float* t_vm    = alloc((size_t)Bc * Mc * 8);
  float* t_msg   = alloc((size_t)Bc * Oc * 8);
  float* op0     = alloc((size_t)Bc * 8);
  float* ma0     = alloc((size_t)Bc * 8);
  float* op_pool = alloc((size_t)Bc * 8);
  float* ma_pool = alloc((size_t)Bc * 8);
  float* scoresb = alloc((size_t)Bc * Mc * Jc);

  norm_opes_kernel<<<Bc, 256, 0, stream>>>(raw_opes, opes_n);
  norm_mas_kernel<<<1, 32, 0, stream>>>(raw_mas, mas_n);
  proc_partial_kernel<<<256, 256, 0, stream>>>(proc, partial);
  proc_final_kernel<<<1, 1, 0, stream>>>(partial, pstats);

  const int gO = (Bc * Oc + 255) / 256;   // 125 blocks of 256 over (b,o)
  const int gM = (Bc * Mc + 255) / 256;   // 3 blocks over (b,m)
  const float* opes_in = opes_n; int din_o = 4;
  const float* mas_in = mas_n;  int din_m = 2;
  for (int i = 0; i < 2; ++i) {
    float* hmas_out = (i == 0) ? hmasA : hmasB;
    gat_proj_opes_kernel<<<gO, 256, 0, stream>>>(opes_in, din_o, gat_Wo[i], gat_al[i], t_ho, t_el);
    gat_proj_mas_kernel<<<gM, 256, 0, stream>>>(mas_in, din_m, gat_Wm[i], gat_ar[i], t_hm, t_er);
    gat_attn_kernel<<<Bc * Mc, 256, 0, stream>>>(t_ho, t_el, t_hm, t_er, proc, pstats,
                                                 adj_om, gat_we[i], gat_ae[i], hmas_out);
    float* hopes_out = (i == 0) ? hopesA : hopesB;
    hgt_proj_opes_kernel<<<gO, 256, 0, stream>>>(opes_in, din_o, hgt_Wq[i], hgt_Wko[i],
                                                 hgt_Wvo[i], hgt_Wskip[i], t_q, t_ko, t_vo, t_skip);
    hgt_proj_mas_kernel<<<gM, 256, 0, stream>>>(hmas_out, hgt_Wkm[i], hgt_Wvm[i], t_km, t_vm);
    hgt_attn_kernel<<<(Bc * Oc * Hc) / 8, 256, 0, stream>>>(t_q, t_km, t_vm, t_ko, t_vo,
                                                            adj_om, adj_oo, t_msg);
    hgt_out_kernel<<<gO, 256, 0, stream>>>(t_msg, t_skip, hgt_Wout[i], hopes_out);
    opes_in = hopes_out; din_o = 8;
    mas_in = hmas_out;  din_m = 8;
  }
  const float* hopes_f = opes_in;
  const float* hmas_f = mas_in;

  pool_opes_kernel<<<Bc, 256, 0, stream>>>(hopes_f, op0);
  pool_mas_kernel<<<1, 256, 0, stream>>>(hmas_f, ma0);
  head_kernel<<<Bc, 64, 0, stream>>>(scale_in, op0, ma0, c1w1, c1w2, pw1, pw2, op_pool, ma_pool);

  action_mlp_kernel<<<(Bc * Mc * Jc) / 16, 128, 0, stream>>>(
      hopes_f, hmas_f, op_pool, ma_pool, jobs, aw1, ab1, aw2, ab2, aw3, ab3, scoresb);

  softmax_kernel<<<Bc, 256, 0, stream>>>(scoresb, elig, (float*)d_out);
}